// BCQ_Attention_LSTM_64716567216557
// MI455X (gfx1250) — compile-verified
//
#include <hip/hip_runtime.h>
#include <math.h>

typedef float v2f __attribute__((ext_vector_type(2)));
typedef float v8f __attribute__((ext_vector_type(8)));

static constexpr int kMCL  = 100;
static constexpr int kQ    = 100;
static constexpr int kIN   = 200;   // INPUT_DIM
static constexpr int kBERT = 768;
static constexpr int kEMB  = 100;
static constexpr int kDF   = 500;   // D_FULL
static constexpr int kTOT  = 1468;  // TOTAL
static constexpr int kB    = 8;
static constexpr int kT    = 200;
static constexpr int kBT   = kB * kT;
static constexpr int kMP   = 112;   // M padded to 7*16
static constexpr int kNP   = 512;   // N padded
static constexpr int kG4H  = 512;   // 4*HIDDEN

// ---------------------------------------------------------------- init pads
__global__ void k_init(const float* __restrict__ W_pt, const float* __restrict__ b_pt,
                       const float* __restrict__ W_att,
                       float* __restrict__ Wpt_pad, float* __restrict__ bpt_pad,
                       float* __restrict__ watt_pad) {
  int i = blockIdx.x * 256 + threadIdx.x;
  if (i < kNP) {
    bpt_pad[i]  = (i < kDF) ? b_pt[i]  : 0.f;
    watt_pad[i] = (i < kDF) ? W_att[i] : 0.f;
  }
  if (i < kDF * kNP) {
    int k = i >> 9, n = i & (kNP - 1);
    Wpt_pad[i] = (n < kDF) ? W_pt[k * kDF + n] : 0.f;
  }
}

// ------------------------------------------------- phase 1: ctx via f32 WMMA
// grid = B*T, block = 256 (8 waves), dynamic LDS = (112*500 + 112)*4 bytes
__global__ void k_ctx(const float* __restrict__ src,
                      const float* __restrict__ emb_nodes,
                      const float* __restrict__ emb_paths,
                      const float* __restrict__ Wpt_pad,   // [500][512]
                      const float* __restrict__ bpt_pad,   // [512]
                      const float* __restrict__ watt_pad,  // [512]
                      const float* __restrict__ b_att,     // [1]
                      float* __restrict__ ctx_out) {       // [B][MCL][T]
  extern __shared__ float smem[];
  float* full = smem;                  // 112 x 500, row-major
  float* ctx  = smem + kMP * kDF;      // 112

  const int bt  = blockIdx.x;
  const int b   = bt / kT, t = bt % kT;
  const int tid = threadIdx.x;
  const float* row = src + (size_t)bt * kTOT;

  // zero pad rows 100..111 and the ctx accumulator
  for (int i = tid; i < 12 * kDF; i += 256) full[kMCL * kDF + i] = 0.f;
  for (int i = tid; i < kMP; i += 256) ctx[i] = 0.f;
  // broadcast rnn_first into cols 300..499 of every clause row
  for (int i = tid; i < kMCL * kIN; i += 256) {
    int m = i / kIN, j = i % kIN;
    full[m * kDF + 300 + j] = row[j];
  }
  // gather embeddings: cols [0,100)=start node, [100,200)=end node, [200,300)=path
  for (int i = tid; i < kMCL * 300; i += 256) {
    int m = i / 300, j = i % 300;
    float v;
    if (j < 100) {
      int idx = (int)row[kIN + 3 * m + 0];
      v = emb_nodes[(size_t)idx * kEMB + j];
    } else if (j < 200) {
      int idx = (int)row[kIN + 3 * m + 2];
      v = emb_nodes[(size_t)idx * kEMB + (j - 100)];
    } else {
      int idx = (int)row[kIN + 3 * m + 1];
      v = emb_paths[(size_t)idx * kEMB + (j - 200)];
    }
    full[m * kDF + j] = v;
  }
  __syncthreads();

  const int wave = tid >> 5;
  const int lane = tid & 31;
  const int l15  = lane & 15;
  const int ksel = lane >> 4;            // 0: lanes 0-15, 1: lanes 16-31

  // per-lane base into the LDS tile: row (l15), col (2*ksel)
  const float* abase = full + (size_t)l15 * kDF + 2 * ksel;

  for (int ntb = 0; ntb < 4; ++ntb) {
    const int nt = wave + ntb * 8;       // this wave's N-tile (0..31)
    const int n  = nt * 16 + l15;        // column handled by this lane

    v8f C[7];
#pragma unroll
    for (int mt = 0; mt < 7; ++mt)
#pragma unroll
      for (int i = 0; i < 8; ++i) C[mt][i] = 0.f;

    // software-pipelined B fragment (global/L2, longest latency)
    const float* bcol = Wpt_pad + (size_t)(2 * ksel) * kNP + n;
    v2f bf;
    bf[0] = bcol[0];
    bf[1] = bcol[kNP];

#pragma unroll 1
    for (int k0 = 0; k0 < kDF; k0 += 4) {
      // prefetch next k-step's B fragment (clamped, branch-free)
      const int kn = (k0 + 4 < kDF) ? (k0 + 4) : 0;
      v2f bfn;
      bfn[0] = bcol[(size_t)kn * kNP];
      bfn[1] = bcol[(size_t)kn * kNP + kNP];

      // batch-load all 7 A fragments so DS loads can clause + partial-wait
      v2f af[7];
#pragma unroll
      for (int mt = 0; mt < 7; ++mt)
        af[mt] = *(const v2f*)(abase + (size_t)mt * 16 * kDF + k0);

#pragma unroll
      for (int mt = 0; mt < 7; ++mt)
        C[mt] = __builtin_amdgcn_wmma_f32_16x16x4_f32(
            false, af[mt], false, bf, (short)0, C[mt], false, false);

      bf = bfn;
    }

    // epilogue per N-tile: tanh, weight, 16-lane reduce, LDS atomic accumulate
    const float bb = bpt_pad[n];
    const float ww = watt_pad[n];
#pragma unroll
    for (int mt = 0; mt < 7; ++mt)
#pragma unroll
      for (int i = 0; i < 8; ++i) {
        float v = tanhf(C[mt][i] + bb) * ww;
        v += __shfl_xor(v, 1, 32);
        v += __shfl_xor(v, 2, 32);
        v += __shfl_xor(v, 4, 32);
        v += __shfl_xor(v, 8, 32);
        const int m = mt * 16 + i + 8 * ksel;   // C row for this half
        if (l15 == 0 && m < kMCL) atomicAdd(&ctx[m], v);
      }
  }
  __syncthreads();

  const float ba = b_att[0];
  for (int m = tid; m < kMCL; m += 256)
    ctx_out[((size_t)b * kMCL + m) * kT + t] = ctx[m] + ba;
}

// ------------------------------------------ phase 2: softmax over T per (b,m)
__global__ void k_softmax_t(float* __restrict__ ctx) {   // grid = B*MCL
  __shared__ float red[256];
  const int tid = threadIdx.x;
  float* p = ctx + (size_t)blockIdx.x * kT;
  float v = (tid < kT) ? p[tid] : -1e30f;
  red[tid] = v; __syncthreads();
  for (int s = 128; s > 0; s >>= 1) {
    if (tid < s) red[tid] = fmaxf(red[tid], red[tid + s]);
    __syncthreads();
  }
  const float mx = red[0]; __syncthreads();
  float e = (tid < kT) ? expf(v - mx) : 0.f;
  red[tid] = e; __syncthreads();
  for (int s = 128; s > 0; s >>= 1) {
    if (tid < s) red[tid] += red[tid + s];
    __syncthreads();
  }
  if (tid < kT) p[tid] = e / red[0];
}

// --- phase 3: per (b,t): code_vec, bert_vec, rnn_full@W_ff, xpre = fused@W_ih
__global__ void k_row(const float* __restrict__ src,
                      const float* __restrict__ emb_nodes,
                      const float* __restrict__ emb_paths,
                      const float* __restrict__ attw,      // [B][MCL][T]
                      const float* __restrict__ W_bert, const float* __restrict__ b_bert,
                      const float* __restrict__ W_ff,  const float* __restrict__ b_ff,
                      const float* __restrict__ W_ih,  const float* __restrict__ b_ih,
                      const float* __restrict__ b_hh,
                      const float* __restrict__ W_q,   const float* __restrict__ b_q,
                      float* __restrict__ xpre,            // [B*T][512]
                      float* __restrict__ wscore) {        // [B*T]
  __shared__ int   sidx[kMCL], pidx[kMCL], eidx[kMCL];
  __shared__ float aw[kMCL], cv[kDF], bert[100], rf[1000], fused[kDF], red[256];
  __shared__ float sum_aw;
  const int bt = blockIdx.x, b = bt / kT, t = bt % kT, tid = threadIdx.x;
  const float* row = src + (size_t)bt * kTOT;

  if (tid < kMCL) {
    sidx[tid] = (int)row[kIN + 3 * tid + 0];
    pidx[tid] = (int)row[kIN + 3 * tid + 1];
    eidx[tid] = (int)row[kIN + 3 * tid + 2];
    aw[tid]   = attw[((size_t)b * kMCL + tid) * kT + t];
  }
  __syncthreads();
  if (tid == 0) {
    float s = 0.f;
    for (int m = 0; m < kMCL; ++m) s += aw[m];
    sum_aw = s;
  }
  for (int j = tid; j < 300; j += 256) {
    float s = 0.f;
    if (j < 100)      for (int m = 0; m < kMCL; ++m) s += aw[m] * emb_nodes[(size_t)sidx[m] * kEMB + j];
    else if (j < 200) for (int m = 0; m < kMCL; ++m) s += aw[m] * emb_nodes[(size_t)eidx[m] * kEMB + (j - 100)];
    else              for (int m = 0; m < kMCL; ++m) s += aw[m] * emb_paths[(size_t)pidx[m] * kEMB + (j - 200)];
    cv[j] = s;
  }
  // bert_vec = bert_id @ W_bert + b_bert
  for (int n = tid; n < 100; n += 256) {
    float s = b_bert[n];
    for (int k = 0; k < kBERT; ++k) s += row[500 + k] * W_bert[k * 100 + n];
    bert[n] = s;
  }
  __syncthreads();                       // sum_aw, cv[0:300), bert ready
  for (int j = tid; j < kIN; j += 256) cv[300 + j] = row[j] * sum_aw;
  __syncthreads();
  // rnn_full = [time(50), score(50), rnn_first(200), code_vec(500), bert(100), kc(100)]
  for (int j = tid; j < 1000; j += 256) {
    float v;
    if (j < 50)       v = row[1268 + j];
    else if (j < 100) v = row[1318 + (j - 50)];
    else if (j < 300) v = row[j - 100];
    else if (j < 800) v = cv[j - 300];
    else if (j < 900) v = bert[j - 800];
    else              v = row[1368 + (j - 900)];
    rf[j] = v;
  }
  __syncthreads();
  // fused = [kc(100), rnn_first(200), rnn_full@W_ff + b_ff (200)]
  for (int n = tid; n < 200; n += 256) {
    float s = b_ff[n];
    for (int k = 0; k < 1000; ++k) s += rf[k] * W_ff[k * 200 + n];
    fused[300 + n] = s;
  }
  for (int j = tid; j < 100; j += 256) fused[j] = row[1368 + j];
  for (int j = tid; j < 200; j += 256) fused[100 + j] = row[j];
  __syncthreads();
  // xpre = fused @ W_ih + (b_ih + b_hh)
  for (int n = tid; n < kG4H; n += 256) {
    float s = b_ih[n] + b_hh[n];
    for (int k = 0; k < kDF; ++k) s += fused[k] * W_ih[k * kG4H + n];
    xpre[(size_t)bt * kG4H + n] = s;
  }
  // score w = [bert_vec, kc] @ W_q + b_q
  float ps = 0.f;
  if (tid < 100)      ps = bert[tid] * W_q[tid];
  else if (tid < 200) ps = row[1368 + (tid - 100)] * W_q[tid];
  red[tid] = ps; __syncthreads();
  for (int s = 128; s > 0; s >>= 1) {
    if (tid < s) red[tid] += red[tid + s];
    __syncthreads();
  }
  if (tid == 0) wscore[bt] = red[0] + b_q[0];
}

// ------------------- phase 4: LSTM, one WG per batch row, W_hh staged in LDS
// grid = 8, block = 512, dynamic LDS = (128*512 + 128 + 512)*4 bytes
__global__ void k_lstm(const float* __restrict__ W_hh,
                       const float* __restrict__ xpre,
                       float* __restrict__ out) {          // [B*T][128]
  extern __shared__ float s[];
  float* whh = s;                 // 128 x 512
  float* h   = s + 128 * kG4H;    // 128
  float* g   = h + 128;           // 512
  const int b = blockIdx.x, tid = threadIdx.x;
  for (int i = tid; i < 128 * kG4H; i += 512) whh[i] = W_hh[i];
  if (tid < 128) h[tid] = 0.f;
  float c = 0.f;
  __syncthreads();
  for (int t = 0; t < kT; ++t) {
    float acc = xpre[((size_t)b * kT + t) * kG4H + tid];
    for (int k = 0; k < 128; ++k) acc += h[k] * whh[k * kG4H + tid];
    g[tid] = acc;
    __syncthreads();
    if (tid < 128) {
      const float i_ = g[tid], f_ = g[128 + tid], gg = g[256 + tid], o_ = g[384 + tid];
      const float si = 1.f / (1.f + expf(-i_));
      const float sf = 1.f / (1.f + expf(-f_));
      const float so = 1.f / (1.f + expf(-o_));
      c = sf * c + si * tanhf(gg);
      const float hh = so * tanhf(c);
      h[tid] = hh;
      out[((size_t)b * kT + t) * 128 + tid] = hh;
    }
    __syncthreads();
  }
}

// ---------------------- phase 5: exp weights + prefix attention, one WG per b
__global__ void k_att(const float* __restrict__ wscore,
                      const float* __restrict__ out,
                      float* __restrict__ att) {
  __shared__ float e[kT];
  __shared__ float red[256];
  const int b = blockIdx.x, tid = threadIdx.x;
  float w = (tid < kT) ? wscore[b * kT + tid] : -1e30f;
  red[tid] = w; __syncthreads();
  for (int s = 128; s > 0; s >>= 1) {
    if (tid < s) red[tid] = fmaxf(red[tid], red[tid + s]);
    __syncthreads();
  }
  const float mx = red[0];
  if (tid < kT) e[tid] = expf(w - mx);
  __syncthreads();
  if (tid < 128) {
    float num = 0.f, den = 0.f;
    for (int t = 0; t < kT; ++t) {
      const float o  = out[((size_t)b * kT + t) * 128 + tid];
      const float et = e[t];
      num += et * o; den += et;
      float a;
      if (t == 0)           a = o;
      else if (t == kT - 1) a = 0.5f * o;
      else                  a = 0.5f * o + 0.5f * (num / den);
      att[((size_t)b * kT + t) * 128 + tid] = a;
    }
  }
}

// ------------------------------- phase 6: final @ W_fc + sigmoid, per (b,t)
__global__ void k_final(const float* __restrict__ src,
                        const float* __restrict__ att,
                        const float* __restrict__ W_fc, const float* __restrict__ b_fc,
                        float* __restrict__ dout) {
  __shared__ float fin[428];
  const int bt = blockIdx.x, tid = threadIdx.x;
  const float* row = src + (size_t)bt * kTOT;
  for (int j = tid; j < 200; j += 128) fin[j] = row[j];
  for (int j = tid; j < 100; j += 128) fin[200 + j] = row[1368 + j];
  if (tid < 128) fin[300 + tid] = att[(size_t)bt * 128 + tid];
  __syncthreads();
  if (tid < kQ) {
    float s = b_fc[tid];
    for (int k = 0; k < 428; ++k) s += fin[k] * W_fc[k * kQ + tid];
    dout[(size_t)bt * kQ + tid] = 1.f / (1.f + expf(-s));
  }
}

// ---------------------------------------------------------------------------
extern "C" void kernel_launch(void* const* d_in, const int* in_sizes, int n_in,
                              void* d_out, int out_size, void* d_ws, size_t ws_size,
                              hipStream_t stream) {
  const float* src       = (const float*)d_in[0];
  const float* emb_nodes = (const float*)d_in[1];
  const float* emb_paths = (const float*)d_in[2];
  const float* W_bert = (const float*)d_in[3];
  const float* b_bert = (const float*)d_in[4];
  const float* W_pt   = (const float*)d_in[5];
  const float* b_pt   = (const float*)d_in[6];
  const float* W_att  = (const float*)d_in[7];
  const float* b_att  = (const float*)d_in[8];
  const float* W_ff   = (const float*)d_in[9];
  const float* b_ff   = (const float*)d_in[10];
  const float* W_ih   = (const float*)d_in[11];
  const float* W_hh   = (const float*)d_in[12];
  const float* b_ih   = (const float*)d_in[13];
  const float* b_hh   = (const float*)d_in[14];
  const float* W_q    = (const float*)d_in[15];
  const float* b_q    = (const float*)d_in[16];
  const float* W_fc   = (const float*)d_in[17];
  const float* b_fc   = (const float*)d_in[18];

  float* ws = (float*)d_ws;
  float* Wpt_pad  = ws;                         // 500*512
  float* bpt_pad  = Wpt_pad + kDF * kNP;        // 512
  float* watt_pad = bpt_pad + kNP;              // 512
  float* ctx      = watt_pad + kNP;             // 8*100*200
  float* xpre     = ctx + kB * kMCL * kT;       // 1600*512
  float* wscore   = xpre + (size_t)kBT * kG4H;  // 1600
  float* lout     = wscore + kBT;               // 1600*128
  float* att      = lout + (size_t)kBT * 128;   // 1600*128

  const int ctx_lds  = (kMP * kDF + kMP) * 4;            // ~224.4 KB
  const int lstm_lds = (128 * kG4H + 128 + kG4H) * 4;    // ~264.7 KB
  (void)hipFuncSetAttribute((const void*)k_ctx,
      hipFuncAttributeMaxDynamicSharedMemorySize, ctx_lds);
  (void)hipFuncSetAttribute((const void*)k_lstm,
      hipFuncAttributeMaxDynamicSharedMemorySize, lstm_lds);

  k_init<<<(kDF * kNP + 255) / 256, 256, 0, stream>>>(W_pt, b_pt, W_att,
                                                      Wpt_pad, bpt_pad, watt_pad);
  k_ctx<<<kBT, 256, ctx_lds, stream>>>(src, emb_nodes, emb_paths,
                                       Wpt_pad, bpt_pad, watt_pad, b_att, ctx);
  k_softmax_t<<<kB * kMCL, 256, 0, stream>>>(ctx);
  k_row<<<kBT, 256, 0, stream>>>(src, emb_nodes, emb_paths, ctx,
                                 W_bert, b_bert, W_ff, b_ff,
                                 W_ih, b_ih, b_hh, W_q, b_q, xpre, wscore);
  k_lstm<<<kB, 512, lstm_lds, stream>>>(W_hh, xpre, lout);
  k_att<<<kB, 256, 0, stream>>>(wscore, lout, att);
  k_final<<<kBT, 128, 0, stream>>>(src, att, W_fc, b_fc, (float*)d_out);
}